// FilterConstructorTree_14250701488165
// MI455X (gfx1250) — compile-verified
//
#include <hip/hip_runtime.h>

#define B_     32
#define T_     1024
#define DIM_   512
#define N_     16
#define DEPTH_ 3

typedef float v2f __attribute__((ext_vector_type(2)));
typedef float v4f __attribute__((ext_vector_type(4)));
typedef float v8f __attribute__((ext_vector_type(8)));

// workspace layout (float offsets)
#define WS_POOLED   0                    // 16384  : pooled state [B][DIM] (mutated per depth)
#define WS_CUMBIAS  16384                // 16384  : cumulative bias [B][DIM]
#define WS_SCALE    32768                // 49152  : s_d [DEPTH][B][DIM]
#define WS_ADD      81920                // 49152  : c_d [DEPTH][B][DIM]
#define WS_PARTIAL  131072               // 524288 : mean partials [B*32][DIM]

// ---------------- Kernel 1a: partial sums over T (32 rows per block) ----------------
__global__ void pooled_stage1(const float* __restrict__ x, float* __restrict__ partial) {
    int blk = blockIdx.x;            // B*32 blocks
    int b = blk >> 5, c = blk & 31;
    int i = threadIdx.x;             // 512 threads = DIM
    const float* xp = x + ((size_t)b * T_ + (size_t)c * 32) * DIM_ + i;
    float s = 0.0f;
#pragma unroll 8
    for (int r = 0; r < 32; ++r) s += xp[(size_t)r * DIM_];
    partial[(size_t)blk * DIM_ + i] = s;
}

// ---------------- Kernel 1b: reduce partials -> pooled mean ----------------
__global__ void pooled_stage2(const float* __restrict__ partial, float* __restrict__ pooled) {
    int b = blockIdx.x;              // B blocks
    int i = threadIdx.x;             // 512 threads
    float s = 0.0f;
#pragma unroll 8
    for (int c = 0; c < 32; ++c) s += partial[((size_t)b * 32 + c) * DIM_ + i];
    pooled[(size_t)b * DIM_ + i] = s * (1.0f / (float)T_);
}

// ---------------- Kernel 2: sequential tree walk (1 block, 256 threads) ----------------
__global__ void tree_walk(const float* __restrict__ kd0, const float* __restrict__ kd1,
                          const float* __restrict__ kd2,
                          const float* __restrict__ vd0, const float* __restrict__ vd1,
                          const float* __restrict__ vd2,
                          const float* __restrict__ W,  const float* __restrict__ gumbel,
                          float* __restrict__ ws) {
    float* pooled = ws + WS_POOLED;
    float* cum    = ws + WS_CUMBIAS;
    float* wsS    = ws + WS_SCALE;
    float* wsA    = ws + WS_ADD;
    const float* kd[DEPTH_] = {kd0, kd1, kd2};
    const float* vd[DEPTH_] = {vd0, vd1, vd2};

    __shared__ float ls_logits[B_ * N_];
    __shared__ float ls_ks[B_ * N_];       // straight-through one-hot rows (WMMA A-matrix)
    __shared__ int   ls_node[B_], ls_nodeNext[B_], ls_hot[B_];
    __shared__ float ls_ksHot[B_], ls_prevkUse[B_], ls_prevkCur[B_];

    const int tid    = threadIdx.x;
    const int wave   = tid >> 5;
    const int lane   = tid & 31;
    const int laneLo = lane & 15;
    const int laneHi = lane >> 4;

    if (tid < B_) { ls_node[tid] = 0; ls_prevkCur[tid] = 1.0f; }
    for (int j = tid; j < B_ * N_;   j += 256) ls_ks[j] = 0.0f;
    for (int j = tid; j < B_ * DIM_; j += 256) cum[j]   = 0.0f;
    __syncthreads();

    for (int d = 0; d < DEPTH_; ++d) {
        // ---- bias update: cum += ks @ W, pooled += ks @ W  (V_WMMA_F32_16X16X4_F32) ----
        if (d > 0) {
            for (int tile = wave; tile < 64; tile += 8) {   // 2 M-tiles x 32 N-tiles
                int mT = tile >> 5, nT = tile & 31;
                v8f acc = {0.f, 0.f, 0.f, 0.f, 0.f, 0.f, 0.f, 0.f};
#pragma unroll
                for (int kk = 0; kk < N_; kk += 4) {
                    v2f a, bm;
                    // A (16x4 f32): lanes 0-15 = rows; VGPR0 holds K=kk+0|kk+2, VGPR1 K=kk+1|kk+3
                    a.x  = ls_ks[(mT * 16 + laneLo) * N_ + kk + laneHi * 2 + 0];
                    a.y  = ls_ks[(mT * 16 + laneLo) * N_ + kk + laneHi * 2 + 1];
                    // B (4x16 f32): lanes 0-15 = cols; same K striping
                    bm.x = W[(size_t)(kk + laneHi * 2 + 0) * DIM_ + nT * 16 + laneLo];
                    bm.y = W[(size_t)(kk + laneHi * 2 + 1) * DIM_ + nT * 16 + laneLo];
                    acc = __builtin_amdgcn_wmma_f32_16x16x4_f32(
                            false, a, false, bm, (short)0, acc, false, false);
                }
#pragma unroll
                for (int r = 0; r < 8; ++r) {   // D: VGPR r = rows M=r (lanes 0-15), M=r+8 (16-31)
                    int bIdx = mT * 16 + r + laneHi * 8;
                    int ii   = nT * 16 + laneLo;
                    float dlt = acc[r];
                    cum[bIdx * DIM_ + ii]    += dlt;
                    pooled[bIdx * DIM_ + ii] += dlt;
                }
            }
            __syncthreads();
        }

        // ---- logits[b][n] = pooled[b] . kd[d][node[b]][n]  (per-sample gathered weights) ----
        const float* kdl = kd[d];
        for (int p = wave; p < B_ * N_; p += 8) {
            int b = p >> 4, n = p & 15;
            const float* krow = kdl + ((size_t)ls_node[b] * N_ + n) * DIM_;
            const float* prow = pooled + (size_t)b * DIM_;
            float s = 0.0f;
            for (int j = lane; j < DIM_; j += 32) s += prow[j] * krow[j];
#pragma unroll
            for (int off = 16; off > 0; off >>= 1) s += __shfl_down(s, off, 32);
            if (lane == 0) ls_logits[p] = s;
        }
        __syncthreads();

        // ---- softmax + gumbel argmax + straight-through one-hot (one lane per sample) ----
        if (tid < B_) {
            int b = tid;
            float l[N_], e[N_];
            float mx = -3.4e38f;
#pragma unroll
            for (int n = 0; n < N_; ++n) { l[n] = ls_logits[b * N_ + n]; mx = fmaxf(mx, l[n]); }
            float sum = 0.0f;
#pragma unroll
            for (int n = 0; n < N_; ++n) { e[n] = expf(l[n] - mx); sum += e[n]; }
            float inv = 1.0f / sum;
            float best = -3.4e38f; int hot = 0;
#pragma unroll
            for (int n = 0; n < N_; ++n) {
                float pr = e[n] * inv;
                float v  = logf(pr) + gumbel[((size_t)d * B_ + b) * N_ + n];
                if (v > best) { best = v; hot = n; }   // first-max tiebreak, like argmax
            }
            float ph  = e[hot] * inv;
            float ksv = (1.0f + ph) - ph;              // replicate onehot + p - p fp semantics
#pragma unroll
            for (int n = 0; n < N_; ++n) ls_ks[b * N_ + n] = (n == hot) ? ksv : 0.0f;
            ls_hot[b] = hot; ls_ksHot[b] = ksv;
            float pk = ls_prevkCur[b];
            ls_prevkUse[b] = pk;                       // values use pre-update prev_keys
            ls_prevkCur[b] = pk * ksv;                 // max(k_sample) == ksv
            ls_nodeNext[b] = ls_node[b] * N_ + hot;
        }
        __syncthreads();

        // ---- emit per-depth coefficient tables: s = sel*prevk ; c = cumBias*s ----
        const float* vdl = vd[d];
        for (int f = tid; f < B_ * DIM_; f += 256) {
            int b = f >> 9, i = f & (DIM_ - 1);
            const float* vrow = vdl + ((size_t)ls_node[b] * N_ + ls_hot[b]) * DIM_;
            float s = ls_ksHot[b] * vrow[i] * ls_prevkUse[b];
            wsS[(size_t)d * B_ * DIM_ + f] = s;
            wsA[(size_t)d * B_ * DIM_ + f] = cum[f] * s;
        }
        __syncthreads();
        if (tid < B_) ls_node[tid] = ls_nodeNext[tid];
        __syncthreads();
    }
}

// ---------------- Kernel 3: out[d] = x * s_d + c_d  (bandwidth-bound emission) ----------------
__global__ void emit(const float* __restrict__ x, const float* __restrict__ wsS,
                     const float* __restrict__ wsA, float* __restrict__ out) {
    __shared__ float sS[DEPTH_][DIM_];
    __shared__ float sA[DEPTH_][DIM_];
    int b  = blockIdx.x >> 6;        // 32 samples
    int tc = blockIdx.x & 63;        // 64 chunks of 16 rows
    for (int j = threadIdx.x; j < DEPTH_ * DIM_; j += 512) {
        int d = j >> 9, i = j & (DIM_ - 1);
        sS[d][i] = wsS[((size_t)d * B_ + b) * DIM_ + i];
        sA[d][i] = wsA[((size_t)d * B_ + b) * DIM_ + i];
    }
    __syncthreads();
    const v4f* xp = (const v4f*)(x + ((size_t)b * T_ + (size_t)tc * 16) * DIM_);
    for (int q = threadIdx.x; q < 16 * 128; q += 512) {
        int r = q >> 7, i4 = q & 127;
        v4f xv = xp[r * 128 + i4];
        int i = i4 * 4;
        size_t base = ((size_t)b * T_ + (size_t)tc * 16 + r) * DIM_ + i;
#pragma unroll
        for (int d = 0; d < DEPTH_; ++d) {
            v4f o;
            o.x = fmaf(xv.x, sS[d][i + 0], sA[d][i + 0]);
            o.y = fmaf(xv.y, sS[d][i + 1], sA[d][i + 1]);
            o.z = fmaf(xv.z, sS[d][i + 2], sA[d][i + 2]);
            o.w = fmaf(xv.w, sS[d][i + 3], sA[d][i + 3]);
            // output never re-read: non-temporal b128 store, keep x resident in L2
            __builtin_nontemporal_store(o, (v4f*)(out + (size_t)d * B_ * T_ * DIM_ + base));
        }
    }
}

extern "C" void kernel_launch(void* const* d_in, const int* in_sizes, int n_in,
                              void* d_out, int out_size, void* d_ws, size_t ws_size,
                              hipStream_t stream) {
    const float* x      = (const float*)d_in[0];
    const float* kd0    = (const float*)d_in[1];
    const float* kd1    = (const float*)d_in[2];
    const float* kd2    = (const float*)d_in[3];
    const float* vd0    = (const float*)d_in[4];
    const float* vd1    = (const float*)d_in[5];
    const float* vd2    = (const float*)d_in[6];
    const float* W      = (const float*)d_in[7];
    const float* gumbel = (const float*)d_in[8];
    float* ws  = (float*)d_ws;
    float* out = (float*)d_out;

    pooled_stage1<<<B_ * 32, DIM_, 0, stream>>>(x, ws + WS_PARTIAL);
    pooled_stage2<<<B_, DIM_, 0, stream>>>(ws + WS_PARTIAL, ws + WS_POOLED);
    tree_walk<<<1, 256, 0, stream>>>(kd0, kd1, kd2, vd0, vd1, vd2, W, gumbel, ws);
    emit<<<B_ * 64, 512, 0, stream>>>(x, ws + WS_SCALE, ws + WS_ADD, out);
}